// MedianAbsoluteDeviation_75548474737335
// MI455X (gfx1250) — compile-verified
//
#include <hip/hip_runtime.h>
#include <cstdint>
#include <cstddef>

// MAD 3x3 filter on channel 1 of an (8,3,512,512) f32 tensor, reflect padding.
// HBM-bound (~16.8 MB traffic); CDNA5 path used: async global->LDS staging
// (ASYNCcnt) + split wait counters. WMMA is inapplicable (order statistics).

#define TILE_W   64
#define TILE_H   4
#define HALO_W   (TILE_W + 2)   // 66
#define HALO_H   (TILE_H + 2)   // 6
#define NTHREADS 256

// ---- CDNA5 async global->LDS copy (per-lane, tracked with ASYNCcnt) -------
__device__ __forceinline__ void async_g2l_b32(const float* g, float* l) {
  // generic -> LDS address-space cast gives a 32-bit LDS byte offset VGPR
  __attribute__((address_space(3))) void* lp =
      (__attribute__((address_space(3))) void*)l;
  asm volatile("global_load_async_to_lds_b32 %0, %1, off"
               :: "v"(lp), "v"(g)
               : "memory");
}

__device__ __forceinline__ void wait_async_then_barrier() {
  asm volatile("s_wait_asynccnt 0x0" ::: "memory");
  __syncthreads();
}

// ---- median-of-9 (Paeth sorting network, 19 min/max pairs) ----------------
__device__ __forceinline__ void sort2(float& a, float& b) {
  float t = fminf(a, b);
  b = fmaxf(a, b);
  a = t;
}

__device__ __forceinline__ float median9(float p0, float p1, float p2,
                                         float p3, float p4, float p5,
                                         float p6, float p7, float p8) {
  sort2(p1, p2); sort2(p4, p5); sort2(p7, p8);
  sort2(p0, p1); sort2(p3, p4); sort2(p6, p7);
  sort2(p1, p2); sort2(p4, p5); sort2(p7, p8);
  sort2(p0, p3); sort2(p5, p8); sort2(p4, p7);
  sort2(p3, p6); sort2(p1, p4); sort2(p2, p5);
  sort2(p4, p7); sort2(p4, p2); sort2(p6, p4);
  sort2(p4, p2);
  return p4;
}

// reflect (no edge repeat): -1 -> 1, n -> n-2  (pad width is exactly 1)
__device__ __forceinline__ int reflect1(int i, int n) {
  if (i < 0)  i = -i;
  if (i >= n) i = 2 * n - 2 - i;
  return i;
}

__global__ __launch_bounds__(NTHREADS)
void MedianAbsoluteDeviation_75548474737335_kernel(const float* __restrict__ x,
                                                   float* __restrict__ out,
                                                   int H, int W, int C, int center)
{
  __shared__ float tile[HALO_H * HALO_W];

  const int b  = blockIdx.z;
  const int h0 = blockIdx.y * TILE_H;
  const int w0 = blockIdx.x * TILE_W;
  const int t  = threadIdx.x;

  const float* xc = x + ((size_t)b * C + center) * (size_t)H * W;

  // Stage 66x6 halo tile into LDS with async copies (reflect folded into addr).
  for (int i = t; i < HALO_H * HALO_W; i += NTHREADS) {
    int r  = i / HALO_W;
    int c  = i - r * HALO_W;
    int gh = reflect1(h0 - 1 + r, H);
    int gw = reflect1(w0 - 1 + c, W);
    async_g2l_b32(xc + (size_t)gh * W + gw, &tile[i]);
  }
  wait_async_then_barrier();

  const int ty = t >> 6;           // t / TILE_W
  const int tx = t & (TILE_W - 1); // t % TILE_W

  const float* rp = &tile[ty * HALO_W + tx];
  float w00 = rp[0],            w01 = rp[1],            w02 = rp[2];
  float w10 = rp[HALO_W + 0],   w11 = rp[HALO_W + 1],   w12 = rp[HALO_W + 2];
  float w20 = rp[2*HALO_W + 0], w21 = rp[2*HALO_W + 1], w22 = rp[2*HALO_W + 2];

  float med = median9(w00, w01, w02, w10, w11, w12, w20, w21, w22);

  float d0 = fabsf(w00 - med), d1 = fabsf(w01 - med), d2 = fabsf(w02 - med);
  float d3 = fabsf(w10 - med), d4 = fabsf(w11 - med), d5 = fabsf(w12 - med);
  float d6 = fabsf(w20 - med), d7 = fabsf(w21 - med), d8 = fabsf(w22 - med);

  float mad = median9(d0, d1, d2, d3, d4, d5, d6, d7, d8);

  out[((size_t)b * H + (h0 + ty)) * (size_t)W + (w0 + tx)] = mad;
}

extern "C" void kernel_launch(void* const* d_in, const int* in_sizes, int n_in,
                              void* d_out, int out_size, void* d_ws, size_t ws_size,
                              hipStream_t stream) {
  (void)in_sizes; (void)n_in; (void)out_size; (void)d_ws; (void)ws_size;

  const float* x   = (const float*)d_in[0];
  float*       out = (float*)d_out;

  const int B = 8, C = 3, H = 512, W = 512;
  const int center = C / 2; // == 1

  dim3 grid(W / TILE_W, H / TILE_H, B); // (8, 128, 8)
  dim3 block(NTHREADS);
  MedianAbsoluteDeviation_75548474737335_kernel<<<grid, block, 0, stream>>>(
      x, out, H, W, C, center);
}